// CentLoss_18356690223602
// MI455X (gfx1250) — compile-verified
//
#include <hip/hip_runtime.h>
#include <math.h>

#define HH 128
#define WW 128
#define CC 80
#define NBOX 64
#define BATCH 16
#define STRIDE_F 8.0f
#define MIN_OVERLAP_F 0.7f
#define TILE_PIX 128
#define THREADS 256
#define IMG_STRIDE 256   // floats per image in workspace

typedef float v2f __attribute__((ext_vector_type(2)));
typedef float v8f __attribute__((ext_vector_type(8)));

// workspace (floats): [0]=acc_kp, [1]=acc_size, per-image block at 16 + b*IMG_STRIDE:
//   [0]=count(int), [1]=inv2sig2, [2..65]=gy, [66..129]=gx, [130..193]=cls0(int)

__global__ void cent_init_kernel(float* ws) {
    if (threadIdx.x < 2) ws[threadIdx.x] = 0.0f;
}

__global__ void cent_prep_kernel(const float* __restrict__ kp,
                                 const float* __restrict__ preg,
                                 const float* __restrict__ gt,
                                 float* ws) {
    int b = blockIdx.x;
    int tid = threadIdx.x;                 // 0..63
    __shared__ float s_y[NBOX], s_x[NBOX], s_tmp[NBOX], s_tmp2[NBOX];
    __shared__ int   s_cls[NBOX], s_yi[NBOX], s_xi[NBOX];
    __shared__ int   s_count;
    __shared__ float s_sigma;

    const float* g = gt + (b * NBOX + tid) * 7;
    float gy0 = g[0], gx0 = g[1];
    float gh = (g[2] + g[4]) * (1.0f / STRIDE_F);
    float gw = (g[3] + g[5]) * (1.0f / STRIDE_F);
    s_y[tid]  = gy0 * (1.0f / STRIDE_F);
    s_x[tid]  = gx0 * (1.0f / STRIDE_F);
    s_cls[tid] = (int)g[6];
    s_yi[tid] = (int)floorf(gy0 * (1.0f / STRIDE_F));
    s_xi[tid] = (int)floorf(gx0 * (1.0f / STRIDE_F));
    s_tmp[tid] = gy0;
    __syncthreads();
    if (tid == 0) {                        // argmin, first occurrence
        int best = 0; float bv = s_tmp[0];
        for (int i = 1; i < NBOX; ++i) if (s_tmp[i] < bv) { bv = s_tmp[i]; best = i; }
        s_count = best;
    }
    __syncthreads();
    int count = s_count;
    bool valid = tid < count;

    // gaussian radius (min of r1,r2,r3), reduced over valid boxes
    float rmin = INFINITY;
    if (valid) {
        const float mo = MIN_OVERLAP_F;
        float h = gh, w = gw;
        float b1 = h + w,            c1 = w * h * (1.0f - mo) / (1.0f + mo);
        float r1 = (b1 + sqrtf(b1 * b1 - 4.0f * c1)) * 0.5f;
        float b2 = 2.0f * (h + w),   c2 = (1.0f - mo) * w * h;
        float r2 = (b2 + sqrtf(b2 * b2 - 16.0f * c2)) * 0.5f;
        float b3 = -2.0f * mo * (h + w), c3 = (mo - 1.0f) * w * h;
        float r3 = (b3 + sqrtf(b3 * b3 - 16.0f * mo * c3)) * 0.5f;
        rmin = fminf(r1, fminf(r2, r3));
    }
    s_tmp[tid] = rmin;
    __syncthreads();
    if (tid == 0) {
        float m = s_tmp[0];
        for (int i = 1; i < NBOX; ++i) m = fminf(m, s_tmp[i]);
        s_sigma = m;
    }
    __syncthreads();
    float sigma = s_sigma;
    float inv2s2 = 1.0f / (2.0f * sigma * sigma);

    float* img = ws + 16 + b * IMG_STRIDE;
    if (tid == 0) { ((int*)img)[0] = count; img[1] = inv2s2; }
    img[2 + tid]  = valid ? s_y[tid] : 1.0e7f;   // padded boxes -> pen==0
    img[66 + tid] = valid ? s_x[tid] : 1.0e7f;
    ((int*)img)[130 + tid] = valid ? (s_cls[tid] - 1) : 0;

    // size loss partial
    float ssum = 0.0f;
    if (valid) {
        int yi = s_yi[tid], xi = s_xi[tid];
        const float* pr = preg + (((size_t)b * HH + yi) * WW + xi) * 4;
        for (int k = 0; k < 4; ++k) {
            float d = g[2 + k] * (1.0f / STRIDE_F) - pr[k];
            ssum += d * d;
        }
    }

    // center correction: + pos term, - neg term that the dense pass will add there
    float corr = 0.0f;
    if (valid) {
        int yi = s_yi[tid], xi = s_xi[tid], c0 = s_cls[tid] - 1;
        bool dup = false;
        for (int j = 0; j < tid; ++j)
            if (s_yi[j] == yi && s_xi[j] == xi && s_cls[j] == s_cls[tid]) { dup = true; break; }
        if (!dup) {
            float red = 0.0f, fy = (float)yi, fx = (float)xi;
            for (int j = 0; j < count; ++j)
                if (s_cls[j] == s_cls[tid]) {
                    float dy = s_y[j] - fy, dx = s_x[j] - fx;
                    red = fmaxf(red, expf(-(dy * dy + dx * dx) * inv2s2));
                }
            float kv = kp[(((size_t)b * HH + yi) * WW + xi) * CC + c0];
            float e  = expf(-fabsf(kv));
            float lg = log1pf(e);
            float ls = (kv >= 0.0f) ? -lg : kv - lg;           // log sigmoid
            float sg = (kv >= 0.0f) ? 1.0f / (1.0f + e) : e / (1.0f + e);
            float l1ms = ls - kv;                              // log(1 - sigmoid)
            float oms = 1.0f - sg;
            float pos = -(oms * oms) * ls;
            float om  = 1.0f - red; float om2 = om * om;
            float neg = -(om2 * om2) * sg * sg * l1ms;
            corr = pos - neg;
        }
    }

    s_tmp[tid]  = ssum;
    s_tmp2[tid] = corr;
    __syncthreads();
    if (tid == 0 && count > 0) {
        float a = 0.0f, c = 0.0f;
        for (int i = 0; i < NBOX; ++i) { a += s_tmp[i]; c += s_tmp2[i]; }
        float ng = (float)count;
        atomicAdd(ws + 1, a / (ng * 4.0f));
        atomicAdd(ws + 0, c / ng);
    }
}

__device__ __forceinline__ float cent_neg_term(float kv, float red) {
    float e   = __expf(-fabsf(kv));
    float lg  = __logf(1.0f + e);
    float ls  = (kv >= 0.0f) ? -lg : kv - lg;        // log sigmoid
    float sg  = (kv >= 0.0f) ? 1.0f / (1.0f + e) : e / (1.0f + e);
    float l1ms = ls - kv;                            // log(1 - sigmoid)
    float om = 1.0f - red; float om2 = om * om;
    return -(om2 * om2) * sg * sg * l1ms;
}

__global__ void cent_main_kernel(const float* __restrict__ kp, float* ws) {
    int b = blockIdx.y;
    int pixBase = blockIdx.x * TILE_PIX;
    int tid = threadIdx.x;

    __shared__ __align__(16) int s_red[TILE_PIX * CC]; // 40 KB peak map (f32 bits >= 0)
    __shared__ float s_gy[NBOX], s_gx[NBOX];
    __shared__ int   s_cls[NBOX];
    __shared__ float s_part[THREADS];

    const float* img = ws + 16 + b * IMG_STRIDE;
    int   count  = ((const int*)img)[0];
    float inv2s2 = img[1];
    if (tid < NBOX) {
        s_gy[tid]  = img[2 + tid];
        s_gx[tid]  = img[66 + tid];
        s_cls[tid] = ((const int*)img)[130 + tid];
    }
    {   // b128 zero-fill of the red tile
        int4* r4 = (int4*)s_red;
        int4 z; z.x = 0; z.y = 0; z.z = 0; z.w = 0;
        for (int i = tid; i < (TILE_PIX * CC) / 4; i += THREADS) r4[i] = z;
    }
    __syncthreads();

    // ---- Phase A: pixel-box squared distances via V_WMMA_F32_16X16X4_F32 ----
    int lane = tid & 31;
    int wave = tid >> 5;
    int m = lane & 15;
    int pixLocalBase = wave * 16;             // 8 waves x 16 pixels = 128
    int pix = pixBase + pixLocalBase + m;
    float fh = (float)(pix >> 7);             // row (gy grid)
    float fw = (float)(pix & 127);            // col (gx grid)
    // A (16x4): row m = [h, w, h^2+w^2, 1]; lanes 0-15 hold K0/K1, 16-31 hold K2/K3
    v2f A;
    if (lane < 16) { A.x = fh; A.y = fw; }
    else           { A.x = fh * fh + fw * fw; A.y = 1.0f; }

    int ngroups = (count + 15) >> 4;
    for (int gI = 0; gI < ngroups; ++gI) {
        int n = gI * 16 + m;
        float gyv = s_gy[n], gxv = s_gx[n];
        // B (4x16): col n = [-2gy, -2gx, 1, gy^2+gx^2]
        v2f Bv;
        if (lane < 16) { Bv.x = -2.0f * gyv; Bv.y = -2.0f * gxv; }
        else           { Bv.x = 1.0f;        Bv.y = gyv * gyv + gxv * gxv; }
        v8f Cz = {0.f, 0.f, 0.f, 0.f, 0.f, 0.f, 0.f, 0.f};
        v8f D = __builtin_amdgcn_wmma_f32_16x16x4_f32(
            false, A, false, Bv, (short)0, Cz, false, false);
        int cls = s_cls[n];
        #pragma unroll
        for (int r = 0; r < 8; ++r) {
            int pl = pixLocalBase + ((lane < 16) ? r : r + 8);
            float pen = __expf(-D[r] * inv2s2);
            atomicMax(&s_red[pl * CC + cls], __float_as_int(pen));
        }
    }
    __syncthreads();

    // ---- Phase B: streaming focal "neg" over the keypoint tile (b128 loads) ----
    const float4* kp4 =
        (const float4*)(kp + ((size_t)b * HH * WW + pixBase) * CC);
    const float4* red4p = (const float4*)s_red;
    float acc = 0.0f;
    for (int i = tid; i < (TILE_PIX * CC) / 4; i += THREADS) {
        float4 kv4 = kp4[i];
        float4 rd4 = red4p[i];
        acc += cent_neg_term(kv4.x, rd4.x);
        acc += cent_neg_term(kv4.y, rd4.y);
        acc += cent_neg_term(kv4.z, rd4.z);
        acc += cent_neg_term(kv4.w, rd4.w);
    }
    s_part[tid] = acc;
    __syncthreads();
    for (int s = THREADS / 2; s > 0; s >>= 1) {
        if (tid < s) s_part[tid] += s_part[tid + s];
        __syncthreads();
    }
    if (tid == 0 && count > 0) atomicAdd(ws + 0, s_part[0] / (float)count);
}

__global__ void cent_finalize_kernel(const float* __restrict__ ws, float* out) {
    out[0] = ws[0] * (1.0f / (float)BATCH) + ws[1] * (0.1f / (float)BATCH);
}

extern "C" void kernel_launch(void* const* d_in, const int* in_sizes, int n_in,
                              void* d_out, int out_size, void* d_ws, size_t ws_size,
                              hipStream_t stream) {
    const float* kp   = (const float*)d_in[0];   // (16,128,128,80)
    const float* preg = (const float*)d_in[1];   // (16,128,128,4)
    const float* gt   = (const float*)d_in[2];   // (16,64,7)
    float* ws = (float*)d_ws;

    hipLaunchKernelGGL(cent_init_kernel, dim3(1), dim3(32), 0, stream, ws);
    hipLaunchKernelGGL(cent_prep_kernel, dim3(BATCH), dim3(NBOX), 0, stream,
                       kp, preg, gt, ws);
    dim3 grid(HH * WW / TILE_PIX, BATCH);
    hipLaunchKernelGGL(cent_main_kernel, grid, dim3(THREADS), 0, stream, kp, ws);
    hipLaunchKernelGGL(cent_finalize_kernel, dim3(1), dim3(1), 0, stream,
                       ws, (float*)d_out);
}